// LinearLayerAttention_19189913879289
// MI455X (gfx1250) — compile-verified
//
#include <hip/hip_runtime.h>

typedef __attribute__((ext_vector_type(2))) float v2f;
typedef __attribute__((ext_vector_type(8))) float v8f;

#define Bc 8
#define Cc 64
#define Hc 56
#define Wc 56
#define HWc (Hc * Wc)      /* 3136  */
#define Dc  (Cc * HWc)     /* 200704 */
#define EPSc 1e-6f

// ---------------------------------------------------------------------------
// 1) y[b,c] = mean over HxW of x[b,c,:,:]
// ---------------------------------------------------------------------------
__global__ void pool_kernel(const float* __restrict__ x, float* __restrict__ y) {
    __shared__ float red[256];
    const int bc = blockIdx.x;                  // b*C + c
    const float* xp = x + (size_t)bc * HWc;
    float sum = 0.f;
    for (int i = threadIdx.x; i < HWc; i += 256) sum += xp[i];
    red[threadIdx.x] = sum;
    __syncthreads();
    for (int off = 128; off > 0; off >>= 1) {
        if ((int)threadIdx.x < off) red[threadIdx.x] += red[threadIdx.x + off];
        __syncthreads();
    }
    if (threadIdx.x == 0) y[bc] = red[0] * (1.0f / (float)HWc);
}

__device__ __forceinline__ float phi_elu1(float u) {
    // elu(u) + 1 : u>0 -> u+1 ; u<=0 -> exp(u)
    return u > 0.f ? u + 1.f : __expf(u);
}

// ---------------------------------------------------------------------------
// 2) Q,K = phi(conv1d_same(y, w)), z_new = z + K, QZ = 1/sum_c Q*(z_new+eps)
// ---------------------------------------------------------------------------
__global__ void qk_kernel(const float* __restrict__ y, const float* __restrict__ z,
                          const float* __restrict__ wq, const float* __restrict__ wk,
                          float* __restrict__ Q, float* __restrict__ K,
                          float* __restrict__ QZ, float* __restrict__ z_new) {
    const int b = blockIdx.x;
    const int c = threadIdx.x;                  // 0..63
    const float* yb = y + b * Cc;
    const float ym1 = (c > 0)      ? yb[c - 1] : 0.f;
    const float y0  = yb[c];
    const float yp1 = (c < Cc - 1) ? yb[c + 1] : 0.f;
    const float qv = phi_elu1(ym1 * wq[0] + y0 * wq[1] + yp1 * wq[2]);
    const float kv = phi_elu1(ym1 * wk[0] + y0 * wk[1] + yp1 * wk[2]);
    const float zn = z[b * Cc + c] + kv;
    Q[b * Cc + c] = qv;
    K[b * Cc + c] = kv;
    z_new[b * Cc + c] = zn;
    __shared__ float prod[Cc];
    prod[c] = qv * (zn + EPSc);
    __syncthreads();
    if (c == 0) {
        float ssum = 0.f;
        for (int i = 0; i < Cc; ++i) ssum += prod[i];
        QZ[b] = 1.0f / ssum;
    }
}

// ---------------------------------------------------------------------------
// 3) V = depthwise 3x3 conv of x ('same' zero padding)
// ---------------------------------------------------------------------------
__global__ void dwconv_kernel(const float* __restrict__ x, const float* __restrict__ wv,
                              float* __restrict__ V) {
    const long n = (long)blockIdx.x * blockDim.x + threadIdx.x;
    if (n >= (long)Bc * Dc) return;
    const int w = (int)(n % Wc);
    const int h = (int)((n / Wc) % Hc);
    const int c = (int)((n / HWc) % Cc);
    const float* xp = x + (n - (long)h * Wc - w);   // x[b,c,0,0]
    const float* wp = wv + c * 9;
    float acc = 0.f;
#pragma unroll
    for (int i = 0; i < 3; ++i) {
        const int hh = h + i - 1;
        if (hh < 0 || hh >= Hc) continue;
#pragma unroll
        for (int j = 0; j < 3; ++j) {
            const int ww = w + j - 1;
            if (ww < 0 || ww >= Wc) continue;
            acc += xp[hh * Wc + ww] * wp[i * 3 + j];
        }
    }
    V[n] = acc;
}

// ---------------------------------------------------------------------------
// 4) Fused streaming pass (bandwidth-critical; ~850 MB total):
//    per 16-column tile:  s_new = s + K*V  (store),  out = QZ * (Q . s_new)
//    The c-reduction is done with V_WMMA_F32_16X16X4_F32:
//      A (16x4 f32): all 16 rows = Q[cb..cb+3]
//        lanes 0-15 : v0=Q[cb+0], v1=Q[cb+1] (K0,K1)
//        lanes 16-31: v0=Q[cb+2], v1=Q[cb+3] (K2,K3)
//      B (4x16 f32): column 'col' per lane pair, same K split
//    With identical A rows, every lane's acc[0] = result for its column.
// ---------------------------------------------------------------------------
__global__ void __launch_bounds__(128)
stream_wmma_kernel(const float* __restrict__ s, const float* __restrict__ V,
                   const float* __restrict__ Q, const float* __restrict__ K,
                   const float* __restrict__ QZ,
                   float* __restrict__ out, float* __restrict__ s_new) {
    const int lane = threadIdx.x & 31;
    const int wid  = threadIdx.x >> 5;
    const int b    = blockIdx.y;
    const long tile = (long)blockIdx.x * 4 + wid;    // one 16-wide d tile per wave
    const long d0   = tile * 16;
    const int col   = lane & 15;
    const int half  = lane >> 4;                     // 0: K0/K1 rows, 1: K2/K3 rows
    const long d    = d0 + col;

    const float* sB  = s     + (size_t)b * Cc * Dc;
    float*       snB = s_new + (size_t)b * Cc * Dc;
    const float vval = V[(size_t)b * Dc + d];
    const float qz   = QZ[b];

    v8f acc = {};
#pragma unroll 4
    for (int cb = 0; cb < Cc; cb += 4) {
        const int rA = cb + 2 * half;                           // even row index
        const float2 qp = *(const float2*)(Q + b * Cc + rA);    // global_load_b64
        const float2 kp = *(const float2*)(K + b * Cc + rA);
        const size_t off0 = (size_t)rA * Dc + (size_t)d;

        // prefetch next c-block's s rows (gfx1250 global_prefetch; speculative)
        __builtin_prefetch(sB + off0 + 4 * (size_t)Dc, 0, 0);

        const float s0 = sB[off0];
        const float s1 = sB[off0 + Dc];
        const float sn0 = fmaf(kp.x, vval, s0);
        const float sn1 = fmaf(kp.y, vval, s1);
        snB[off0]      = sn0;
        snB[off0 + Dc] = sn1;

        v2f a;  a.x = qp.x;  a.y = qp.y;     // A fragment (Q broadcast rows)
        v2f bm; bm.x = sn0;  bm.y = sn1;     // B fragment (s_new tile)
        acc = __builtin_amdgcn_wmma_f32_16x16x4_f32(
            /*neg_a=*/false, a, /*neg_b=*/false, bm,
            /*c_mod=*/(short)0, acc, /*reuse_a=*/false, /*reuse_b=*/false);
    }
    if (half == 0) out[(size_t)b * Dc + d] = acc[0] * qz;
}

// ---------------------------------------------------------------------------
extern "C" void kernel_launch(void* const* d_in, const int* in_sizes, int n_in,
                              void* d_out, int out_size, void* d_ws, size_t ws_size,
                              hipStream_t stream) {
    const float* x  = (const float*)d_in[0];
    const float* s  = (const float*)d_in[1];
    const float* z  = (const float*)d_in[2];
    const float* wq = (const float*)d_in[3];
    const float* wk = (const float*)d_in[4];
    const float* wv = (const float*)d_in[5];

    // d_out = [ out : B*D | s_new : B*C*D | z_new : B*C ]
    float* out   = (float*)d_out;
    float* s_new = out + (size_t)Bc * Dc;
    float* z_new = s_new + (size_t)Bc * Cc * Dc;

    // workspace: y, Q, K, QZ, V
    float* ws = (float*)d_ws;
    float* y  = ws;
    float* Q  = y  + Bc * Cc;
    float* K  = Q  + Bc * Cc;
    float* QZ = K  + Bc * Cc;
    float* V  = QZ + Bc;                     // B*D floats

    pool_kernel<<<Bc * Cc, 256, 0, stream>>>(x, y);
    qk_kernel<<<Bc, Cc, 0, stream>>>(y, z, wq, wk, Q, K, QZ, z_new);

    const long nV = (long)Bc * Dc;
    dwconv_kernel<<<(int)((nV + 255) / 256), 256, 0, stream>>>(x, wv, V);

    dim3 grid(Dc / (16 * 4), Bc);            // 3136 x 8 blocks, 4 waves/block
    stream_wmma_kernel<<<grid, 128, 0, stream>>>(s, V, Q, K, QZ, out, s_new);
}